// ProbAttention_61899068670718
// MI455X (gfx1250) — compile-verified
//
#include <hip/hip_runtime.h>
#include <math.h>

// ProbSparse attention (Informer) for MI455X / gfx1250, fp32 end-to-end.
// Bandwidth-bound (~95 MB moved, ~1.1 GFLOP): use V_WMMA_F32_16X16X4_F32 for
// the two dense GEMMs (scores = Qr@K^T, context = attn@V) to stay exact in f32.

typedef __attribute__((ext_vector_type(2))) float vf2;
typedef __attribute__((ext_vector_type(8))) float vf8;

#define BB 2
#define HH 8
#define LQ 4096
#define SS 4096
#define DD 64
#define UU 45
#define UP 48   // 45 padded to 3 WMMA M-tiles

// ---------------------------------------------------------------------------
// K1: M[b,h,l] = max_u(q . k_idx[l,u]) - sum_u(q . k_idx[l,u]) / S
// One wave32 per query; lane i holds dims (2i, 2i+1); xor-shuffle reduce.
// ---------------------------------------------------------------------------
__global__ void k_sample_m(const float* __restrict__ Q,
                           const float* __restrict__ K,
                           const int* __restrict__ idx,
                           float* __restrict__ M) {
  int wid = (blockIdx.x * blockDim.x + threadIdx.x) >> 5;   // global wave id
  int lane = threadIdx.x & 31;
  int l = wid & (LQ - 1);
  int bh = wid >> 12;           // wid / LQ
  int b = bh >> 3, h = bh & 7;

  const float* qp = Q + (((size_t)b * LQ + l) * HH + h) * DD;
  vf2 q = *(const vf2*)(qp + 2 * lane);

  float mx = -INFINITY, sm = 0.0f;
  for (int u = 0; u < UU; ++u) {
    int s = idx[l * UU + u];
    const float* kp = K + (((size_t)b * SS + s) * HH + h) * DD;
    vf2 kv = *(const vf2*)(kp + 2 * lane);
    float p = q.x * kv.x + q.y * kv.y;
    for (int off = 16; off > 0; off >>= 1) p += __shfl_xor(p, off, 32);
    mx = fmaxf(mx, p);
    sm += p;
  }
  if (lane == 0) M[wid] = mx - sm / (float)SS;
}

// ---------------------------------------------------------------------------
// K2: per (b,h): iterative top-45 of M over L, ties -> lower index
// (matches jax.lax.top_k ordering).
// ---------------------------------------------------------------------------
__global__ void k_topk(const float* __restrict__ M, int* __restrict__ mtop) {
  __shared__ float vals[LQ];
  __shared__ float rv[256];
  __shared__ int ri[256];
  int bh = blockIdx.x, t = threadIdx.x;

  for (int j = t; j < LQ; j += 256) vals[j] = M[(size_t)bh * LQ + j];
  __syncthreads();

  for (int it = 0; it < UU; ++it) {
    float bv = -INFINITY; int bi = 0x7fffffff;
    for (int j = t; j < LQ; j += 256) {       // ascending j: ties keep lower idx
      float v = vals[j];
      if (v > bv) { bv = v; bi = j; }
    }
    rv[t] = bv; ri[t] = bi;
    __syncthreads();
    for (int s = 128; s > 0; s >>= 1) {
      if (t < s) {
        float v2 = rv[t + s]; int i2 = ri[t + s];
        if (v2 > rv[t] || (v2 == rv[t] && i2 < ri[t])) { rv[t] = v2; ri[t] = i2; }
      }
      __syncthreads();
    }
    if (t == 0) { mtop[bh * UU + it] = ri[0]; vals[ri[0]] = -INFINITY; }
    __syncthreads();
  }
}

// ---------------------------------------------------------------------------
// K3: raw scaled scores = (Qr @ K^T) / 8 via V_WMMA_F32_16X16X4_F32.
// Block = (bh, 128-key chunk), 256 threads = 8 waves; wave w -> 16 keys,
// 3 M-tiles (48 padded query rows), K-loop of 16 (D=64, K=4 per wmma).
// A frag (16x4): lane L holds m=L&15, K=(L>>4)*2+{0,1}; B mirrored; ISA 7.12.2.
// ---------------------------------------------------------------------------
__global__ void k_scores(const float* __restrict__ Q,
                         const float* __restrict__ K,
                         const int* __restrict__ mtop,
                         float* __restrict__ attn) {
  __shared__ float qr[UP * DD];                 // 12 KB, zero-padded rows 45..47
  int t = threadIdx.x;
  int bh = blockIdx.x >> 5;                     // 32 chunks of 128 keys per bh
  int chunk = blockIdx.x & 31;
  int b = bh >> 3, h = bh & 7;

  for (int e = t; e < UP * DD; e += 256) {
    int u = e >> 6, d = e & 63;
    float v = 0.0f;
    if (u < UU) {
      int l = mtop[bh * UU + u];
      v = Q[(((size_t)b * LQ + l) * HH + h) * DD + d];
    }
    qr[e] = v;
  }
  __syncthreads();

  int wave = t >> 5, lane = t & 31;
  int s0 = chunk * 128 + wave * 16;
  int nlo = lane & 15;
  int khi = (lane >> 4) * 2;

  vf8 c0 = {}, c1 = {}, c2 = {};
  for (int k = 0; k < 16; ++k) {
    int kd = 4 * k + khi;
    const float* kp = K + (((size_t)b * SS + (s0 + nlo)) * HH + h) * DD + kd;
    vf2 bf = *(const vf2*)kp;                   // key (col) nlo, dims kd,kd+1
    vf2 a0 = *(const vf2*)&qr[(0  + nlo) * DD + kd];
    vf2 a1 = *(const vf2*)&qr[(16 + nlo) * DD + kd];
    vf2 a2 = *(const vf2*)&qr[(32 + nlo) * DD + kd];
    c0 = __builtin_amdgcn_wmma_f32_16x16x4_f32(false, a0, false, bf, (short)0, c0, false, false);
    c1 = __builtin_amdgcn_wmma_f32_16x16x4_f32(false, a1, false, bf, (short)0, c1, false, false);
    c2 = __builtin_amdgcn_wmma_f32_16x16x4_f32(false, a2, false, bf, (short)0, c2, false, false);
  }

  const float scale = 0.125f;                   // 1/sqrt(64)
  size_t base = (size_t)bh * UU * SS;
  int scol = s0 + nlo;
  for (int r = 0; r < 8; ++r) {
    int mrow = r + (lane >> 4) * 8;             // C layout: m = r + 8*(L>>4)
    float v;
    v = c0[r] * scale; if (mrow      < UU) attn[base + (size_t)(mrow     ) * SS + scol] = v;
    v = c1[r] * scale; if (mrow + 16 < UU) attn[base + (size_t)(mrow + 16) * SS + scol] = v;
    v = c2[r] * scale; if (mrow + 32 < UU) attn[base + (size_t)(mrow + 32) * SS + scol] = v;
  }
}

// ---------------------------------------------------------------------------
// K4: per (bh, row): row max, exp in place, row sum -> 1/sum.
// ---------------------------------------------------------------------------
__global__ void k_softmax(float* __restrict__ attn, float* __restrict__ rowinv) {
  __shared__ float red[256];
  int bhu = blockIdx.x, t = threadIdx.x;
  float* row = attn + (size_t)bhu * SS;

  float mx = -INFINITY;
  for (int j = t; j < SS; j += 256) mx = fmaxf(mx, row[j]);
  red[t] = mx; __syncthreads();
  for (int s = 128; s > 0; s >>= 1) { if (t < s) red[t] = fmaxf(red[t], red[t + s]); __syncthreads(); }
  mx = red[0]; __syncthreads();

  float sm = 0.0f;
  for (int j = t; j < SS; j += 256) { float e = expf(row[j] - mx); row[j] = e; sm += e; }
  red[t] = sm; __syncthreads();
  for (int s = 128; s > 0; s >>= 1) { if (t < s) red[t] += red[t + s]; __syncthreads(); }
  if (t == 0) {
    int bh = bhu / UU, u = bhu % UU;
    rowinv[bh * UP + u] = 1.0f / red[0];
  }
}

// ---------------------------------------------------------------------------
// K5: normalize attn in place (final output) and accumulate context partials
// via WMMA. Block = (bh, 512-key chunk), 384 threads = 12 waves; wave ->
// one 16x16 C tile (3 M-tiles x 4 D-tiles). Sub-chunks of 128 keys staged
// in LDS (attn 48x128 = 24 KB, V 128x64 = 32 KB).
// ---------------------------------------------------------------------------
__global__ void k_context(const float* __restrict__ V,
                          float* __restrict__ attn,
                          const float* __restrict__ rowinv,
                          float* __restrict__ ctxpart) {
  __shared__ float at[UP * 128];
  __shared__ float vt[128 * DD];
  int t = threadIdx.x;
  int bh = blockIdx.x >> 3;
  int chunk = blockIdx.x & 7;
  int b = bh >> 3, h = bh & 7;
  int wave = t >> 5, lane = t & 31;
  int mtile = wave >> 2, ntile = wave & 3;
  int nlo = lane & 15;
  int khi = (lane >> 4) * 2;
  size_t abase = (size_t)bh * UU * SS;

  vf8 c = {};
  for (int sc = 0; sc < 4; ++sc) {
    int s0 = chunk * 512 + sc * 128;
    for (int e = t; e < UP * 128; e += 384) {
      int u = e >> 7, cc = e & 127;
      float v = 0.0f;
      if (u < UU) {
        float x = attn[abase + (size_t)u * SS + s0 + cc] * rowinv[bh * UP + u];
        attn[abase + (size_t)u * SS + s0 + cc] = x;   // final normalized attn
        v = x;
      }
      at[e] = v;
    }
    for (int e = t; e < 128 * DD; e += 384) {
      int sr = e >> 6, d = e & 63;
      vt[e] = V[(((size_t)b * SS + s0 + sr) * HH + h) * DD + d];
    }
    __syncthreads();

    for (int k = 0; k < 32; ++k) {
      int sl = 4 * k + khi;
      vf2 a = *(const vf2*)&at[(mtile * 16 + nlo) * 128 + sl];
      vf2 bf;
      bf.x = vt[(sl    ) * DD + ntile * 16 + nlo];
      bf.y = vt[(sl + 1) * DD + ntile * 16 + nlo];
      c = __builtin_amdgcn_wmma_f32_16x16x4_f32(false, a, false, bf, (short)0, c, false, false);
    }
    __syncthreads();
  }

  float* pp = ctxpart + (size_t)(bh * 8 + chunk) * UP * DD;
  for (int r = 0; r < 8; ++r) {
    int u = mtile * 16 + r + (lane >> 4) * 8;
    int d = ntile * 16 + nlo;
    pp[u * DD + d] = c[r];
  }
}

// ---------------------------------------------------------------------------
// K6: reduce 8 chunk partials -> context output (rows 0..44 only).
// ---------------------------------------------------------------------------
__global__ void k_ctx_reduce(const float* __restrict__ ctxpart,
                             float* __restrict__ ctx) {
  int bh = blockIdx.x, t = threadIdx.x;
  for (int e = t; e < UU * DD; e += 256) {
    int u = e >> 6, d = e & 63;
    float s = 0.0f;
    for (int ch = 0; ch < 8; ++ch)
      s += ctxpart[((size_t)(bh * 8 + ch) * UP + u) * DD + d];
    ctx[((size_t)bh * UU + u) * DD + d] = s;
  }
}

// ---------------------------------------------------------------------------
extern "C" void kernel_launch(void* const* d_in, const int* in_sizes, int n_in,
                              void* d_out, int out_size, void* d_ws, size_t ws_size,
                              hipStream_t stream) {
  const float* Q   = (const float*)d_in[0];   // (B,L,H,D)
  const float* K   = (const float*)d_in[1];   // (B,S,H,D)
  const float* V   = (const float*)d_in[2];   // (B,S,H,D)
  const int*   idx = (const int*)d_in[3];     // (L,45)

  float* ctx  = (float*)d_out;                      // (B,H,45,64) = 46080
  float* attn = (float*)d_out + BB * HH * UU * DD;  // (B,H,45,4096)

  // workspace layout (floats)
  float* wsf     = (float*)d_ws;
  float* M       = wsf;                         // 65536
  int*   mtop    = (int*)(wsf + 65536);         // 720 (pad to 768)
  float* rowinv  = wsf + 65536 + 768;           // 16*48
  float* ctxpart = rowinv + 16 * UP;            // 16*8*48*64

  k_sample_m <<<BB * HH * LQ / 8, 256, 0, stream>>>(Q, K, idx, M);
  k_topk     <<<BB * HH,          256, 0, stream>>>(M, mtop);
  k_scores   <<<BB * HH * 32,     256, 0, stream>>>(Q, K, mtop, attn);
  k_softmax  <<<BB * HH * UU,     256, 0, stream>>>(attn, rowinv);
  k_context  <<<BB * HH * 8,      384, 0, stream>>>(V, attn, rowinv, ctxpart);
  k_ctx_reduce<<<BB * HH,         256, 0, stream>>>(ctxpart, ctx);
}